// TokenClusterInter_91130616086565
// MI455X (gfx1250) — compile-verified
//
#include <hip/hip_runtime.h>

// Requires ws_size >= 160*512*512*4 bytes (160 MiB) for the pairwise-distance matrix D.

typedef __attribute__((ext_vector_type(2))) float v2f;
typedef __attribute__((ext_vector_type(8))) float v8f;

#define NBATCH   160
#define NPTS     512
#define NDIM     256
#define KCLUST   49
#define ITERS    15
#define NTHREADS 512
#define NWAVES   (NTHREADS / 32)

#define SPAN     64                 // super-tile edge (rows/cols of D per panel)
#define NSUP     (NPTS / SPAN)      // 8 super-rows / super-cols
#define LDW      260                // LDS row stride in floats (pad: 260 % 64 == 4, 260 % 4 == 0)

__global__ __launch_bounds__(NTHREADS)
void kmedoids_kernel(const float* __restrict__ x,
                     float* __restrict__ out,
                     float* __restrict__ Dws)
{
    const int b = blockIdx.x;
    const float* z = x + (size_t)b * NPTS * NDIM;
    float* Dm = Dws + (size_t)b * NPTS * NPTS;

    __shared__ float s_A[SPAN * LDW];     // 66,560 B  A row-panel (64 rows x 256 cols, padded)
    __shared__ float s_B[SPAN * LDW];     // 66,560 B  B col-panel
    __shared__ float s_sq[NPTS];
    __shared__ float s_cost[NPTS];
    __shared__ int   s_assign[NPTS];
    __shared__ int   s_med[KCLUST];
    __shared__ int   s_newmed[KCLUST];
    __shared__ int   s_done;

    const int tid  = threadIdx.x;
    const int lane = tid & 31;
    const int wave = tid >> 5;

    // ---- row squared norms -> LDS ----
    for (int r = tid; r < NPTS; r += NTHREADS) {
        const float* row = z + r * NDIM;
        float acc = 0.f;
        for (int c = 0; c < NDIM; c += 4) {
            float4 v = *(const float4*)(row + c);
            acc += v.x * v.x + v.y * v.y + v.z * v.z + v.w * v.w;
        }
        s_sq[r] = acc;
    }
    __syncthreads();

    // ---- Phase A: D[i][j] = sqrt(max(sq_i + sq_j - 2*dot, 0)) via fp32 WMMA ----
    // Super-tiled: stage 64-row A panel + 64-col B panel in LDS, 16 waves each
    // compute one 16x16 tile of the 64x64 super-tile.
    const int mrow = lane & 15;
    const int koff = (lane >> 4) * 2;
    const int tm2  = wave >> 2;           // tile row within super-tile (0..3)
    const int tn2  = wave & 3;            // tile col within super-tile (0..3)

    for (int sr = 0; sr < NSUP; ++sr) {
        // stage A panel: z rows [sr*64, sr*64+64), 64*256 floats, float4-coalesced
        for (int q = tid; q < SPAN * (NDIM / 4); q += NTHREADS) {
            const int r  = q >> 6;               // 64 float4 per row
            const int c4 = q & 63;
            float4 v = *(const float4*)(z + (size_t)(sr * SPAN + r) * NDIM + c4 * 4);
            *(float4*)(s_A + r * LDW + c4 * 4) = v;
        }
        for (int sc = 0; sc < NSUP; ++sc) {
            // stage B panel: z rows [sc*64, sc*64+64) (these are D columns)
            for (int q = tid; q < SPAN * (NDIM / 4); q += NTHREADS) {
                const int r  = q >> 6;
                const int c4 = q & 63;
                float4 v = *(const float4*)(z + (size_t)(sc * SPAN + r) * NDIM + c4 * 4);
                *(float4*)(s_B + r * LDW + c4 * 4) = v;
            }
            __syncthreads();

            const int rbase = sr * SPAN + tm2 * 16;
            const int cbase = sc * SPAN + tn2 * 16;
            const float* pA = s_A + (tm2 * 16 + mrow) * LDW + koff;
            const float* pB = s_B + (tn2 * 16 + mrow) * LDW + koff;
            v8f acc = {};
            #pragma unroll 8
            for (int k0 = 0; k0 < NDIM; k0 += 4) {
                v2f a  = *(const v2f*)(pA + k0);
                v2f bb = *(const v2f*)(pB + k0);
                // 8 args: (neg_a, A, neg_b, B, c_mod, C, reuse_a, reuse_b)
                acc = __builtin_amdgcn_wmma_f32_16x16x4_f32(
                    false, a, false, bb, (short)0, acc, false, false);
            }
            // C/D layout: lane n = lane&15, row m = v + 8*(lane>>4)
            const int n  = lane & 15;
            const int mh = (lane >> 4) * 8;
            const float sqn = s_sq[cbase + n];
            #pragma unroll
            for (int v = 0; v < 8; ++v) {
                const int m = mh + v;
                float d2 = s_sq[rbase + m] + sqn - 2.0f * acc[v];
                Dm[(size_t)(rbase + m) * NPTS + (cbase + n)] = sqrtf(fmaxf(d2, 0.0f));
            }
            __syncthreads();   // before restaging s_B (or s_A on next sr)
        }
    }

    // init medoids: med[j] = j*N/k
    for (int j = tid; j < KCLUST; j += NTHREADS) s_med[j] = (j * NPTS) / KCLUST;
    if (tid == 0) s_done = 0;
    __threadfence();     // make this block's D stores visible to its own loads
    __syncthreads();

    // ---- Phase B: k-medoids iterations ----
    for (int it = 0; it < ITERS; ++it) {
        // B1: assign[n] = argmin_j D[n][med[j]]   (tie -> lowest j)
        for (int nn = tid; nn < NPTS; nn += NTHREADS) {
            const float* drow = Dm + (size_t)nn * NPTS;
            float best = 3.402823e38f;
            int   bj = 0;
            for (int j = 0; j < KCLUST; ++j) {
                float d = drow[s_med[j]];
                if (d < best) { best = d; bj = j; }
            }
            s_assign[nn] = bj;
        }
        __syncthreads();

        // B2: cost[i] = sum_{n: assign[n]==assign[i]} D[i][n]  (one wave per row)
        for (int i = wave; i < NPTS; i += NWAVES) {
            const float* drow = Dm + (size_t)i * NPTS;
            const int cj = s_assign[i];
            float acc = 0.f;
            for (int nn = lane; nn < NPTS; nn += 32) {
                acc += (s_assign[nn] == cj) ? drow[nn] : 0.0f;
            }
            #pragma unroll
            for (int off = 16; off > 0; off >>= 1)
                acc += __shfl_down(acc, off, 32);
            if (lane == 0) s_cost[i] = acc;
        }
        __syncthreads();

        // B3: new_med[j] = argmin_{i: assign[i]==j} cost[i]  (tie -> lowest i; empty -> 0)
        for (int j = tid; j < KCLUST; j += NTHREADS) {
            float best = 0.f;
            int   bi = 0;
            int   found = 0;
            for (int i = 0; i < NPTS; ++i) {
                if (s_assign[i] == j) {
                    float c = s_cost[i];
                    if (!found || c < best) { best = c; bi = i; found = 1; }
                }
            }
            s_newmed[j] = bi;
        }
        __syncthreads();

        // B4: sort new_med ascending, convergence test
        if (tid == 0) {
            for (int a = 1; a < KCLUST; ++a) {
                int v = s_newmed[a];
                int p = a - 1;
                while (p >= 0 && s_newmed[p] > v) { s_newmed[p + 1] = s_newmed[p]; --p; }
                s_newmed[p + 1] = v;
            }
            int same = 1;
            for (int j = 0; j < KCLUST; ++j) same &= (s_newmed[j] == s_med[j]);
            s_done = same;
        }
        __syncthreads();
        for (int j = tid; j < KCLUST; j += NTHREADS) s_med[j] = s_newmed[j];
        __syncthreads();
        if (s_done) break;
    }

    // ---- Phase C: out[b][j][:] = z[med[j]][:] ----
    for (int t = tid; t < KCLUST * NDIM; t += NTHREADS) {
        const int j = t >> 8;        // NDIM == 256
        const int w = t & (NDIM - 1);
        out[((size_t)b * KCLUST + j) * NDIM + w] = z[(size_t)s_med[j] * NDIM + w];
    }
}

extern "C" void kernel_launch(void* const* d_in, const int* in_sizes, int n_in,
                              void* d_out, int out_size, void* d_ws, size_t ws_size,
                              hipStream_t stream) {
    (void)in_sizes; (void)n_in; (void)out_size; (void)ws_size;
    const float* x = (const float*)d_in[0];
    float* out = (float*)d_out;
    float* Dws = (float*)d_ws;
    kmedoids_kernel<<<dim3(NBATCH), dim3(NTHREADS), 0, stream>>>(x, out, Dws);
}